// BackProjector_27556510171656
// MI455X (gfx1250) — compile-verified
//
#include <hip/hip_runtime.h>
#include <cstdint>

// ---------------- problem constants (from reference) ----------------
#define ORI   128
#define DIMX  130            // ORI + PF
#define DIMY  259            // DIMX*2 - 1
#define DIMZ  259
#define NIMG  128
#define HH    128            // H
#define WW    65             // W = ORI/2 + 1
#define SLAB  (DIMY * DIMX)          // 33670
#define NVOX  (DIMZ * DIMY * DIMX)   // 8,720,530
#define MAXR2 16384.0                // (ORI/2 * PF)^2

// ---------------- CDNA5 feature probes ----------------
#if defined(__has_builtin)
#  if __has_builtin(__builtin_amdgcn_global_load_async_to_lds_b64)
#    define HAVE_ASYNC_LDS 1
#  endif
#  if __has_builtin(__builtin_amdgcn_s_wait_asynccnt)
#    define HAVE_WAIT_ASYNC 1
#  endif
#endif

// b64 async payload = int2; builtin wants AS1 (global) src, AS3 (LDS) dst
typedef int v2i __attribute__((ext_vector_type(2)));
typedef __attribute__((address_space(1))) v2i gv2i;   // global int2
typedef __attribute__((address_space(3))) v2i lv2i;   // LDS int2

// Vectorized zero-fill: b128 stores to run the init phase at peak store BW.
__global__ void zero_f32_vec(float4* __restrict__ p4, int n4,
                             float* __restrict__ tail, int ntail) {
  int i = blockIdx.x * blockDim.x + threadIdx.x;
  int stride = gridDim.x * blockDim.x;
  const float4 z = make_float4(0.f, 0.f, 0.f, 0.f);
  for (int j = i; j < n4; j += stride) p4[j] = z;
  if (i < ntail) tail[i] = 0.f;     // at most 3 lanes
}

__global__ __launch_bounds__(96) void backproject_kernel(
    const double* __restrict__ f2dr, const double* __restrict__ f2di,
    const double* __restrict__ Amat, const double* __restrict__ Mwt,
    float* __restrict__ out)
{
  const int r = blockIdx.x;     // row in [0,128)
  const int n = blockIdx.y;     // image in [0,128)
  const int t = threadIdx.x;    // 96 threads (3 wave32); cols 0..64 active

  __shared__ double sRe[WW];
  __shared__ double sIm[WW];
  __shared__ double sMw[WW];

  const size_t rowbase = ((size_t)n * HH + (size_t)r) * (size_t)WW;

#if defined(HAVE_ASYNC_LDS)
  if (t < WW) {
    // CDNA5 async copy: global -> LDS via ASYNCcnt (no VGPR data path)
    __builtin_amdgcn_global_load_async_to_lds_b64(
        (gv2i*)(uintptr_t)(f2dr + rowbase + t),
        (lv2i*)(uint32_t)(uintptr_t)&sRe[t], 0, 0);
    __builtin_amdgcn_global_load_async_to_lds_b64(
        (gv2i*)(uintptr_t)(f2di + rowbase + t),
        (lv2i*)(uint32_t)(uintptr_t)&sIm[t], 0, 0);
    __builtin_amdgcn_global_load_async_to_lds_b64(
        (gv2i*)(uintptr_t)(Mwt + rowbase + t),
        (lv2i*)(uint32_t)(uintptr_t)&sMw[t], 0, 0);
  }
#  if defined(HAVE_WAIT_ASYNC)
  __builtin_amdgcn_s_wait_asynccnt(0);
#  else
  asm volatile("s_wait_asynccnt 0" ::: "memory");
#  endif
#else
  if (t < WW) {
    sRe[t] = f2dr[rowbase + t];
    sIm[t] = f2di[rowbase + t];
    sMw[t] = Mwt[rowbase + t];
  }
#endif
  __syncthreads();

  // ---------- per-row geometry (uniform across the block, f64) ----------
  const double* Ab = Amat + (size_t)n * 9;
  // Ainv = A^T * PF; Am = first two cols of Ainv -> Am[k][0]=2*A(0,k), Am[k][1]=2*A(1,k)
  const double m0x = 2.0 * Ab[0], m1x = 2.0 * Ab[1], m2x = 2.0 * Ab[2];
  const double m0y = 2.0 * Ab[3], m1y = 2.0 * Ab[4], m2y = 2.0 * Ab[5];

  const double AtA_xx = m0x * m0x + m1x * m1x + m2x * m2x;
  const double AtA_xy = m0x * m0y + m1x * m1y + m2x * m2y;
  const double AtA_yy = m0y * m0y + m1y * m1y + m2y * m2y;

  // Aflip: p_i = Am[2-i][1]*y + Am[2-i][0]*x
  const double cy0 = m2y, cy1 = m1y, cy2 = m0y;
  const double cx0 = m2x, cx1 = m1x, cx2 = m0x;

  const double y  = (r < WW) ? (double)r : (double)(r - HH);
  const double y2 = y * y;

  const double discr = AtA_xy * AtA_xy * y2 - AtA_xx * (AtA_yy * y2 - MAXR2);
  if (discr < 0.0) return;                       // whole row masked (uniform)

  const double q0 = sqrt(discr) / AtA_xx;
  const double q1 = -AtA_xy * y / AtA_xx;

  double fx = ceil(q1 - q0);
  if (fx < 0.0) fx = 0.0;
  if (r >= WW && fx < 1.0) fx = 1.0;
  double lx = floor(q1 + q0);
  if (lx > (double)(WW - 1)) lx = (double)(WW - 1);

  // ---------- per-pixel work ----------
  if (t >= WW) return;
  const double xd = (double)t;
  if (!(xd >= fx && xd <= lx)) return;

  const double mw = sMw[t];
  if (!(mw > 0.0)) return;

  double p0 = cy0 * y + cx0 * xd;   // z component
  double p1 = cy1 * y + cx1 * xd;   // y component
  double p2 = cy2 * y + cx2 * xd;   // x component

  const double r2 = p0 * p0 + p1 * p1 + p2 * p2;
  if (r2 > MAXR2) return;

  const bool neg = (p2 < 0.0);      // Friedel conjugate half
  if (neg) { p0 = -p0; p1 = -p1; p2 = -p2; }

  const double re = sRe[t];
  double       im = sIm[t];
  if (neg) im = -im;                // conj(f2d)

  const double f0 = floor(p0), f1 = floor(p1), f2 = floor(p2);
  const int iz = (int)f0 + (DIMX - 1);   // init_coords = [1-DIMX, 1-DIMX, 0]
  const int iy = (int)f1 + (DIMX - 1);
  const int ix = (int)f2;
  if (iz < 0 || iy < 0 || ix < 0 || iz >= DIMZ || iy >= DIMY || ix >= DIMX) return;

  const double gz = p0 - f0, gy = p1 - f1, gx = p2 - f2;
  const double wz[2] = {1.0 - gz, gz};
  const double wy[2] = {1.0 - gy, gy};
  const double wx[2] = {1.0 - gx, gx};

  const int base = iz * SLAB + iy * DIMX + ix;
  const float vr = (float)re, vi = (float)im, vm = (float)mw;

  float* __restrict__ oR = out;
  float* __restrict__ oI = out + NVOX;
  float* __restrict__ oW = out + 2 * NVOX;

#pragma unroll
  for (int dz = 0; dz < 2; ++dz) {
#pragma unroll
    for (int dy = 0; dy < 2; ++dy) {
#pragma unroll
      for (int dx = 0; dx < 2; ++dx) {
        const int idx = base + dz * SLAB + dy * DIMX + dx;
        if (idx >= NVOX) continue;   // replicate JAX flat-scatter drop semantics
        const float w = (float)(wz[dz] * wy[dy] * wx[dx]);
        // hardware global_atomic_add_f32 (confirmed: no CAS scaffolding in histogram)
        unsafeAtomicAdd(oR + idx, w * vr);
        unsafeAtomicAdd(oI + idx, w * vi);
        unsafeAtomicAdd(oW + idx, w * vm);
      }
    }
  }
}

extern "C" void kernel_launch(void* const* d_in, const int* in_sizes, int n_in,
                              void* d_out, int out_size, void* d_ws, size_t ws_size,
                              hipStream_t stream) {
  (void)in_sizes; (void)n_in; (void)d_ws; (void)ws_size;
  // setup_inputs() order: f2d_real, f2d_imag, A, Mweight  (all float64)
  const double* f2dr = (const double*)d_in[0];
  const double* f2di = (const double*)d_in[1];
  const double* Amat = (const double*)d_in[2];
  const double* Mwt  = (const double*)d_in[3];
  float* out = (float*)d_out;

  // Output is accumulated with atomics -> must be zeroed every call.
  const int n4    = out_size >> 2;            // float4 chunks (b128 stores)
  const int ntail = out_size & 3;             // trailing floats
  const int zthreads = 256;
  int zblocks = (n4 + zthreads - 1) / zthreads;
  if (zblocks < 1) zblocks = 1;
  zero_f32_vec<<<zblocks, zthreads, 0, stream>>>(
      (float4*)out, n4, out + ((size_t)n4 << 2), ntail);

  dim3 grid(HH, NIMG);            // one block per (row, image)
  backproject_kernel<<<grid, 96, 0, stream>>>(f2dr, f2di, Amat, Mwt, out);
}